// MeasureDistance_58712202936329
// MI455X (gfx1250) — compile-verified
//
#include <hip/hip_runtime.h>
#include <math.h>

typedef __attribute__((ext_vector_type(2))) float v2f;
typedef __attribute__((ext_vector_type(8))) float v8f;

#define EPS_   1.0f
#define B_     8
#define L_     2048      // == K side
#define D_     64
#define NITER_ 10
#define NR_    (B_ * L_) // rows per matrix side = 16384
#define NBG_   2         // batches per L2-resident group (4*NBG_*16MB = 128MB < 192MB L2)
#define NGRP_  (B_ / NBG_)

// ---------------------------------------------------------------------------
// Prep: squared norms of x and y rows
// ---------------------------------------------------------------------------
__global__ void __launch_bounds__(256)
norms_k(const float* __restrict__ x, const float* __restrict__ y,
        float* __restrict__ nx, float* __restrict__ ny) {
    int t = blockIdx.x * 256 + threadIdx.x;          // 0 .. 2*NR_-1
    const float* src;
    float* dst;
    int idx;
    if (t < NR_) { src = x; dst = nx; idx = t; }
    else         { src = y; dst = ny; idx = t - NR_; }
    const float4* p = (const float4*)(src + (size_t)idx * D_);
    float s = 0.f;
#pragma unroll
    for (int i = 0; i < D_ / 4; ++i) {
        float4 v = p[i];
        s += v.x * v.x + v.y * v.y + v.z * v.z + v.w * v.w;
    }
    dst[idx] = s;
}

__global__ void __launch_bounds__(256)
logab_k(const float* __restrict__ a, const float* __restrict__ b,
        float* __restrict__ la, float* __restrict__ lb) {
    int t = blockIdx.x * 256 + threadIdx.x;
    if (t < NR_) la[t] = __logf(a[t]);
    else         lb[t - NR_] = __logf(b[t - NR_]);
}

__global__ void __launch_bounds__(256)
zero_k(float* __restrict__ p, int n) {
    int t = blockIdx.x * 256 + threadIdx.x;
    if (t < n) p[t] = 0.f;
}

// ---------------------------------------------------------------------------
// Cost-matrix build: K[b,i,j] = min(0, 2*u_i.v_j - |u_i|^2 - |v_j|^2)
// One workgroup (8 waves) = 16(M) x 128(N) tile; each wave owns a 16x16 tile
// accumulated with v_wmma_f32_16x16x4_f32 over D=64 (16 WMMAs).
// Optionally also writes the transposed matrix (Kyx = Kxy^T).
// ---------------------------------------------------------------------------
__global__ void __launch_bounds__(256)
kermat_k(const float* __restrict__ U, const float* __restrict__ V,
         const float* __restrict__ nu, const float* __restrict__ nv,
         float* __restrict__ Ko, float* __restrict__ KoT, int batch0) {
    __shared__ float Ul[16 * 68];    // padded stride 68 -> conflict-free
    __shared__ float Vl[128 * 68];

    const int b  = batch0 + blockIdx.z;
    const int m0 = blockIdx.y * 16;
    const int n0 = blockIdx.x * 128;
    const int tid = threadIdx.x;

    const size_t ubase = ((size_t)b * L_ + m0) * D_;
    const size_t vbase = ((size_t)b * L_ + n0) * D_;

    {   // U tile: 16 rows x 64 = 256 float4, one per thread
        int row = tid >> 4, q = tid & 15;
        float4 v = *(const float4*)(U + ubase + (size_t)row * D_ + q * 4);
        *(float4*)&Ul[row * 68 + q * 4] = v;
    }
#pragma unroll
    for (int rep = 0; rep < 8; ++rep) {   // V tile: 128 rows, 8 float4/thread
        int row = rep * 16 + (tid >> 4), q = tid & 15;
        float4 v = *(const float4*)(V + vbase + (size_t)row * D_ + q * 4);
        *(float4*)&Vl[row * 68 + q * 4] = v;
    }
    __syncthreads();

    const int wave = tid >> 5, lane = tid & 31;
    const int half = lane >> 4, r = lane & 15;
    const float* urow = &Ul[r * 68 + 2 * half];
    const float* vrow = &Vl[(wave * 16 + r) * 68 + 2 * half];

    v8f c = {};   // zero accumulator
#pragma unroll
    for (int k0 = 0; k0 < D_; k0 += 4) {
        v2f av = *(const v2f*)(urow + k0);
        v2f bv = *(const v2f*)(vrow + k0);
        c = __builtin_amdgcn_wmma_f32_16x16x4_f32(false, av, false, bv,
                                                  (short)0, c, false, false);
    }

    // Epilogue: C/D layout -> lane half selects M in {vr, vr+8}+m0, N = n0 + wave*16 + r
    const int col = n0 + wave * 16 + r;
    const float nvv = nv[(size_t)b * L_ + col];
    float kv[8];
#pragma unroll
    for (int vr = 0; vr < 8; ++vr) {
        int row = m0 + 8 * half + vr;
        float nuv = nu[(size_t)b * L_ + row];
        float val = fminf(0.f, 2.f * c[vr] - nuv - nvv);
        kv[vr] = val;
        Ko[((size_t)b * L_ + row) * L_ + col] = val;
    }
    if (KoT) {
        size_t tb = ((size_t)b * L_ + col) * L_ + m0 + 8 * half;
        *(float4*)&KoT[tb]     = make_float4(kv[0], kv[1], kv[2], kv[3]);
        *(float4*)&KoT[tb + 4] = make_float4(kv[4], kv[5], kv[6], kv[7]);
    }
}

// ---------------------------------------------------------------------------
// Fused 4-matrix extrapolation over one L2-resident batch group:
//   out = -eps*logsumexp((K+pot)/eps + logw, cols), optional 0.5*(..+potAvg)
// One wave per row; each lane caches 64 row values in VGPRs (16 float4 loads),
// then exact max + exp-sum with wave32 shfl_xor reductions.
// Grid: 4 matrices * NBG_*2048 rows / 8 rows-per-block = 2048 blocks.
// ---------------------------------------------------------------------------
struct ExtrapArgs {
    const float* K[4];
    const float* potExt[4];
    const float* potAvg[4];
    const float* logw[4];
    float*       out[4];
    int          avg;
    int          batch0;
};

__global__ void __launch_bounds__(256)
extrap_k(ExtrapArgs args) {
    const int wave = threadIdx.x >> 5, lane = threadIdx.x & 31;
    const int mat = blockIdx.x >> 9;                        // 512 blocks / matrix
    const int local = ((blockIdx.x & 511) << 3) | wave;     // 0 .. NBG_*2048-1
    const int b = args.batch0 + (local >> 11);
    const int rowInMat = (b << 11) | (local & 2047);

    const float4* krow = (const float4*)(args.K[mat] + (size_t)rowInMat * L_);
    const float4* pe   = (const float4*)(args.potExt[mat] + b * L_);
    const float4* lw   = (const float4*)(args.logw[mat] + b * L_);

    float t[64];
#pragma unroll
    for (int i = 0; i < 16; ++i) {
        int j = i * 32 + lane;           // 0..511 float4 per row
        float4 kvv = krow[j];
        float4 pv  = pe[j];
        float4 lv  = lw[j];
        t[4 * i + 0] = kvv.x + pv.x + lv.x;
        t[4 * i + 1] = kvv.y + pv.y + lv.y;
        t[4 * i + 2] = kvv.z + pv.z + lv.z;
        t[4 * i + 3] = kvv.w + pv.w + lv.w;
    }

    float m = t[0];
#pragma unroll
    for (int k = 1; k < 64; ++k) m = fmaxf(m, t[k]);
#pragma unroll
    for (int off = 16; off >= 1; off >>= 1)
        m = fmaxf(m, __shfl_xor(m, off, 32));

    float s = 0.f;
#pragma unroll
    for (int k = 0; k < 64; ++k) s += __expf(t[k] - m);
#pragma unroll
    for (int off = 16; off >= 1; off >>= 1)
        s += __shfl_xor(s, off, 32);

    if (lane == 0) {
        float o = -EPS_ * (m + __logf(s));
        if (args.avg) o = 0.5f * (o + args.potAvg[mat][rowInMat]);
        args.out[mat][rowInMat] = o;
    }
}

// ---------------------------------------------------------------------------
// Final weighted reduction -> scalar mean over batch
// ---------------------------------------------------------------------------
__global__ void __launch_bounds__(256)
reduce_k(const float* __restrict__ f, const float* __restrict__ g,
         const float* __restrict__ ge, const float* __restrict__ fe,
         const float* __restrict__ a, const float* __restrict__ b,
         float* __restrict__ out) {
    __shared__ float sm[256];
    float acc = 0.f;
    for (int i = threadIdx.x; i < NR_; i += 256) {
        float av = a[i];
        if (av != 0.f) acc += (ge[i] - f[i]) * av;
        float bv = b[i];
        if (bv != 0.f) acc += (fe[i] - g[i]) * bv;
    }
    sm[threadIdx.x] = acc;
    __syncthreads();
    for (int off = 128; off > 0; off >>= 1) {
        if (threadIdx.x < off) sm[threadIdx.x] += sm[threadIdx.x + off];
        __syncthreads();
    }
    if (threadIdx.x == 0) out[0] = sm[0] / (float)B_;
}

// ---------------------------------------------------------------------------
extern "C" void kernel_launch(void* const* d_in, const int* in_sizes, int n_in,
                              void* d_out, int out_size, void* d_ws, size_t ws_size,
                              hipStream_t stream) {
    const float* x = (const float*)d_in[0];  // (B,L,D)
    const float* a = (const float*)d_in[1];  // (B,L)
    const float* y = (const float*)d_in[2];  // (B,K,D)
    const float* b = (const float*)d_in[3];  // (B,K)

    const size_t MAT = (size_t)B_ * L_ * L_;            // 33,554,432 floats
    const size_t need = (4 * MAT + 13 * (size_t)NR_) * sizeof(float);
    if (ws_size < need) return;                          // ws too small: no-op

    float* ws   = (float*)d_ws;
    float* Kxx  = ws;
    float* Kyy  = Kxx + MAT;
    float* Kxy  = Kyy + MAT;
    float* Kyx  = Kxy + MAT;
    float* nx   = Kyx + MAT;
    float* ny   = nx + NR_;
    float* la   = ny + NR_;
    float* lb   = la + NR_;
    float* potA = lb + NR_;          // [pxx, pyy, pxy, pyx]
    float* potB = potA + 4 * NR_;
    float* fin  = potB + 4 * NR_;    // [f, g, ge, fe]

    norms_k<<<2 * NR_ / 256, 256, 0, stream>>>(x, y, nx, ny);
    logab_k<<<2 * NR_ / 256, 256, 0, stream>>>(a, b, la, lb);
    zero_k <<<8 * NR_ / 256, 256, 0, stream>>>(potA, 8 * NR_);  // potA+potB

    const int XBLK = 4 * NBG_ * L_ / 8;   // extrap blocks per group = 2048

    // Process batches in groups of NBG_: the group's 4 cost matrices
    // (4 * NBG_ * 16MB = 128MB) stay resident in the 192MB L2 across the
    // build + 10 iterations + final pass, so the Sinkhorn loop reads L2,
    // not HBM.
    for (int g = 0; g < NGRP_; ++g) {
        const int b0 = g * NBG_;
        dim3 gg(L_ / 128, L_ / 16, NBG_);
        kermat_k<<<gg, 256, 0, stream>>>(x, x, nx, nx, Kxx, nullptr, b0);
        kermat_k<<<gg, 256, 0, stream>>>(y, y, ny, ny, Kyy, nullptr, b0);
        kermat_k<<<gg, 256, 0, stream>>>(x, y, nx, ny, Kxy, Kyx, b0);

        for (int it = 0; it < NITER_; ++it) {
            float* pin  = (it & 1) ? potB : potA;
            float* pout = (it & 1) ? potA : potB;
            ExtrapArgs ea;
            ea.K[0] = Kxx; ea.K[1] = Kyy; ea.K[2] = Kxy; ea.K[3] = Kyx;
            // nxx(pxx), nyy(pyy), nxy uses pyx, nyx uses pxy
            ea.potExt[0] = pin + 0 * NR_; ea.potExt[1] = pin + 1 * NR_;
            ea.potExt[2] = pin + 3 * NR_; ea.potExt[3] = pin + 2 * NR_;
            ea.potAvg[0] = pin + 0 * NR_; ea.potAvg[1] = pin + 1 * NR_;
            ea.potAvg[2] = pin + 2 * NR_; ea.potAvg[3] = pin + 3 * NR_;
            ea.logw[0] = la; ea.logw[1] = lb; ea.logw[2] = lb; ea.logw[3] = la;
            ea.out[0] = pout + 0 * NR_; ea.out[1] = pout + 1 * NR_;
            ea.out[2] = pout + 2 * NR_; ea.out[3] = pout + 3 * NR_;
            ea.avg = 1;
            ea.batch0 = b0;
            extrap_k<<<XBLK, 256, 0, stream>>>(ea);
        }

        {   // final differentiable extrapolation for this group (no averaging)
            float* pin = potA;               // NITER_ even -> latest pots in potA
            ExtrapArgs ea;
            ea.K[0] = Kxx; ea.K[1] = Kyy; ea.K[2] = Kxy; ea.K[3] = Kyx;
            ea.potExt[0] = pin + 0 * NR_;    // f  = extrap(pxx, la, Kxx)
            ea.potExt[1] = pin + 1 * NR_;    // g  = extrap(pyy, lb, Kyy)
            ea.potExt[2] = pin + 3 * NR_;    // ge = extrap(pyx, lb, Kxy)
            ea.potExt[3] = pin + 2 * NR_;    // fe = extrap(pxy, la, Kyx)
            ea.potAvg[0] = ea.potAvg[1] = ea.potAvg[2] = ea.potAvg[3] = pin;
            ea.logw[0] = la; ea.logw[1] = lb; ea.logw[2] = lb; ea.logw[3] = la;
            ea.out[0] = fin + 0 * NR_; ea.out[1] = fin + 1 * NR_;
            ea.out[2] = fin + 2 * NR_; ea.out[3] = fin + 3 * NR_;
            ea.avg = 0;
            ea.batch0 = b0;
            extrap_k<<<XBLK, 256, 0, stream>>>(ea);
        }
    }

    reduce_k<<<1, 256, 0, stream>>>(fin, fin + NR_, fin + 2 * NR_, fin + 3 * NR_,
                                    a, b, (float*)d_out);
}